// GAT_12661563588624
// MI455X (gfx1250) — compile-verified
//
#include <hip/hip_runtime.h>
#include <hip/hip_bf16.h>
#include <math.h>

// ---------------------------------------------------------------------------
// GAT (3-layer) for MI455X / gfx1250.
// Dense transforms via V_WMMA_F32_16X16X4_F32; edge softmax-aggregate via
// three atomic passes (memory/L2-atomic bound part of the roofline).
// ---------------------------------------------------------------------------

typedef __attribute__((ext_vector_type(2))) float v2f;
typedef __attribute__((ext_vector_type(8))) float v8f;

#define GAT_N      100000
#define GAT_HC     64          // heads*out_ch == 64 for all three layers
#define NEG_SLOPE  0.2f

// ---------------------------------------------------------------------------
// GEMM: H[N,64] = A[N,K] @ W[K,64]   (no bias; bias is added post-aggregation)
// Block = 128 threads (4 waves). Wave w computes columns [16w, 16w+16) of a
// 16-row tile. K stepped by 4 with V_WMMA_F32_16X16X4_F32.
//
// f32 WMMA VGPR layouts (ISA 7.12.2):
//   A 16x4 : lanes0-15 {V0=K0,V1=K1}, lanes16-31 {V0=K2,V1=K3}, M = lane&15
//   B 4x16 : lanes0-15 {V0=K0,V1=K1}, lanes16-31 {V0=K2,V1=K3}, N = lane&15
//   C/D    : VGPR r -> M = r + 8*(lane>=16), N = lane&15
// ---------------------------------------------------------------------------
__global__ void gat_gemm_wmma(const float* __restrict__ A,
                              const float* __restrict__ W,
                              float* __restrict__ H, int K) {
  const int tile  = blockIdx.x;            // 16-row tile index
  const int wave  = threadIdx.x >> 5;      // 0..3 -> 16-column block
  const int lane  = threadIdx.x & 31;
  const int half  = lane >> 4;             // 0: lanes 0-15, 1: lanes 16-31
  const int l16   = lane & 15;
  const int rowA  = tile * 16 + l16;       // A row held by this lane
  const int colB  = wave * 16 + l16;       // B/D column held by this lane
  const int kaOff = half * 2;              // K sub-offset for this half-wave

  const float* __restrict__ Arow = A + (size_t)rowA * K;

  v8f acc = {};
  for (int k0 = 0; k0 < K; k0 += 4) {
    // A fragment: two consecutive K values -> single 8B load
    v2f a = *(const v2f*)(Arow + k0 + kaOff);
    // B fragment: W is row-major [K,64]
    v2f b;
    b.x = W[(size_t)(k0 + kaOff) * GAT_HC + colB];
    b.y = W[(size_t)(k0 + kaOff + 1) * GAT_HC + colB];
    acc = __builtin_amdgcn_wmma_f32_16x16x4_f32(
        /*neg_a=*/false, a, /*neg_b=*/false, b,
        /*c_mod=*/(short)0, acc, /*reuse_a=*/false, /*reuse_b=*/false);
  }

#pragma unroll
  for (int r = 0; r < 8; ++r) {
    const int m = half * 8 + r;
    H[(size_t)(tile * 16 + m) * GAT_HC + colB] = acc[r];
  }
}

// ---------------------------------------------------------------------------
// Per-node attention logits: out[n,hd] = dot(h[n, hd*out_ch : ...], avec[hd])
// ---------------------------------------------------------------------------
__global__ void gat_logits(const float* __restrict__ h,
                           const float* __restrict__ avec,
                           float* __restrict__ out,
                           int n_nodes, int heads, int out_ch) {
  const int idx = blockIdx.x * blockDim.x + threadIdx.x;
  if (idx >= n_nodes * heads) return;
  const int n  = idx / heads;
  const int hd = idx - n * heads;
  const float* __restrict__ hp = h + (size_t)n * GAT_HC + hd * out_ch;
  const float* __restrict__ ap = avec + hd * out_ch;
  float s = 0.f;
#pragma unroll 8
  for (int c = 0; c < out_ch; ++c) s = fmaf(hp[c], ap[c], s);
  out[idx] = s;
}

// ---------------------------------------------------------------------------
// Init: agg = 0, m = -inf, denom = 0
// ---------------------------------------------------------------------------
__global__ void gat_init(float* __restrict__ agg, float* __restrict__ m,
                         float* __restrict__ den, int n_agg, int n_head) {
  const int i = blockIdx.x * blockDim.x + threadIdx.x;
  if (i < n_agg) agg[i] = 0.f;
  if (i < n_head) { m[i] = -INFINITY; den[i] = 0.f; }
}

// Float atomic max via signed/unsigned integer monotonicity trick.
__device__ __forceinline__ void atomicMaxFloat(float* addr, float val) {
  if (val >= 0.f)
    atomicMax((int*)addr, __float_as_int(val));
  else
    atomicMin((unsigned int*)addr, __float_as_uint(val));
}

__device__ __forceinline__ void edge_endpoints(const int* __restrict__ ei,
                                               int e, int E, int& src, int& dst) {
  if (e < E) { src = ei[e]; dst = ei[E + e]; }
  else       { src = e - E; dst = e - E; }      // appended self-loop
}

// ---------------------------------------------------------------------------
// Pass A: leaky-relu edge score, atomic segment-max into m[dst,hd]
// ---------------------------------------------------------------------------
__global__ void gat_edge_max(const int* __restrict__ ei, int E, int Etot, int heads,
                             const float* __restrict__ als,
                             const float* __restrict__ ald,
                             float* __restrict__ m) {
  const int idx = blockIdx.x * blockDim.x + threadIdx.x;
  if (idx >= Etot * heads) return;
  const int e  = idx / heads;
  const int hd = idx - e * heads;
  int src, dst; edge_endpoints(ei, e, E, src, dst);
  float v = als[src * heads + hd] + ald[dst * heads + hd];
  v = (v >= 0.f) ? v : NEG_SLOPE * v;
  atomicMaxFloat(&m[dst * heads + hd], v);
}

// ---------------------------------------------------------------------------
// Pass B: ee = exp(score - m[dst]); atomic segment-sum into denom[dst,hd]
// ---------------------------------------------------------------------------
__global__ void gat_edge_exp(const int* __restrict__ ei, int E, int Etot, int heads,
                             const float* __restrict__ als,
                             const float* __restrict__ ald,
                             const float* __restrict__ m,
                             float* __restrict__ ee,
                             float* __restrict__ den) {
  const int idx = blockIdx.x * blockDim.x + threadIdx.x;
  if (idx >= Etot * heads) return;
  const int e  = idx / heads;
  const int hd = idx - e * heads;
  int src, dst; edge_endpoints(ei, e, E, src, dst);
  float v = als[src * heads + hd] + ald[dst * heads + hd];
  v = (v >= 0.f) ? v : NEG_SLOPE * v;
  const float x = __expf(v - m[dst * heads + hd]);
  ee[idx] = x;
  atomicAdd(&den[dst * heads + hd], x);
}

// ---------------------------------------------------------------------------
// Pass C: alpha = ee/denom[dst]; agg[dst, c0..c0+3] += alpha * h[src, c0..c0+3]
// 16 threads per edge, float4 gather, 4x GLOBAL_ATOMIC_ADD_F32 (L2-resident).
// ---------------------------------------------------------------------------
__global__ void gat_edge_scatter(const int* __restrict__ ei, int E, int Etot,
                                 int heads, int out_ch,
                                 const float* __restrict__ h,
                                 const float* __restrict__ ee,
                                 const float* __restrict__ den,
                                 float* __restrict__ agg) {
  const int idx = blockIdx.x * blockDim.x + threadIdx.x;
  if (idx >= Etot * 16) return;
  const int e  = idx >> 4;
  const int c0 = (idx & 15) * 4;
  const int hd = c0 / out_ch;                 // constant over the 4 channels
  int src, dst; edge_endpoints(ei, e, E, src, dst);
  const float alpha = ee[e * heads + hd] / den[dst * heads + hd];
  const float4 hv = *(const float4*)(h + (size_t)src * GAT_HC + c0);
  float* ap = agg + (size_t)dst * GAT_HC + c0;
  atomicAdd(ap + 0, alpha * hv.x);
  atomicAdd(ap + 1, alpha * hv.y);
  atomicAdd(ap + 2, alpha * hv.z);
  atomicAdd(ap + 3, alpha * hv.w);
}

// ---------------------------------------------------------------------------
// Finalize: out = agg + bias (heads=1 mean == identity); optional ReLU
// ---------------------------------------------------------------------------
__global__ void gat_finalize(const float* __restrict__ agg,
                             const float* __restrict__ bias,
                             float* __restrict__ out, int total, int relu) {
  const int i = blockIdx.x * blockDim.x + threadIdx.x;
  if (i >= total) return;
  float v = agg[i] + bias[i % GAT_HC];
  if (relu) v = fmaxf(v, 0.f);
  out[i] = v;
}

// ---------------------------------------------------------------------------
// Host-side per-layer launcher
// ---------------------------------------------------------------------------
static void run_gat_layer(const float* in, int K,
                          const float* W, const float* avs, const float* avd,
                          const float* bias, int heads, int out_ch, int relu,
                          float* out, const int* ei, int E, int Etot,
                          float* bufH, float* agg, float* als, float* ald,
                          float* mmax, float* den, float* ee,
                          hipStream_t stream) {
  const int TB = 256;
  // 1) dense transform (WMMA)
  gat_gemm_wmma<<<GAT_N / 16, 128, 0, stream>>>(in, W, bufH, K);
  // 2) per-node logits
  {
    const int n = GAT_N * heads;
    gat_logits<<<(n + TB - 1) / TB, TB, 0, stream>>>(bufH, avs, als, GAT_N, heads, out_ch);
    gat_logits<<<(n + TB - 1) / TB, TB, 0, stream>>>(bufH, avd, ald, GAT_N, heads, out_ch);
  }
  // 3) init accumulators
  {
    const int n = GAT_N * GAT_HC;
    gat_init<<<(n + TB - 1) / TB, TB, 0, stream>>>(agg, mmax, den, n, GAT_N * heads);
  }
  // 4) segment max
  {
    const int n = Etot * heads;
    gat_edge_max<<<(n + TB - 1) / TB, TB, 0, stream>>>(ei, E, Etot, heads, als, ald, mmax);
  }
  // 5) exp + denom
  {
    const int n = Etot * heads;
    gat_edge_exp<<<(n + TB - 1) / TB, TB, 0, stream>>>(ei, E, Etot, heads, als, ald, mmax, ee, den);
  }
  // 6) weighted scatter
  {
    const int n = Etot * 16;
    gat_edge_scatter<<<(n + TB - 1) / TB, TB, 0, stream>>>(ei, E, Etot, heads, out_ch,
                                                           bufH, ee, den, agg);
  }
  // 7) bias (+relu)
  {
    const int n = GAT_N * GAT_HC;
    gat_finalize<<<(n + TB - 1) / TB, TB, 0, stream>>>(agg, bias, out, n, relu);
  }
}

extern "C" void kernel_launch(void* const* d_in, const int* in_sizes, int n_in,
                              void* d_out, int out_size, void* d_ws, size_t ws_size,
                              hipStream_t stream) {
  // setup_inputs() order:
  //  0:x 1:edge_index 2:W1 3:as1 4:ad1 5:b1 6:W2 7:as2 8:ad2 9:b2
  //  10:W3 11:as3 12:ad3 13:b3
  const float* x   = (const float*)d_in[0];
  const int*   ei  = (const int*)d_in[1];
  const float* W1  = (const float*)d_in[2];
  const float* as1 = (const float*)d_in[3];
  const float* ad1 = (const float*)d_in[4];
  const float* b1  = (const float*)d_in[5];
  const float* W2  = (const float*)d_in[6];
  const float* as2 = (const float*)d_in[7];
  const float* ad2 = (const float*)d_in[8];
  const float* b2  = (const float*)d_in[9];
  const float* W3  = (const float*)d_in[10];
  const float* as3 = (const float*)d_in[11];
  const float* ad3 = (const float*)d_in[12];
  const float* b3  = (const float*)d_in[13];

  const int E    = in_sizes[1] / 2;   // 1,600,000
  const int Etot = E + GAT_N;         // + self loops

  // Workspace carve-up (floats)
  float* ws   = (float*)d_ws;
  float* bufH = ws;                                 // N*64: GEMM output h
  float* act  = bufH + (size_t)GAT_N * GAT_HC;      // N*64: activated layer out
  float* agg  = act  + (size_t)GAT_N * GAT_HC;      // N*64: scatter accumulator
  float* als  = agg  + (size_t)GAT_N * GAT_HC;      // N*2
  float* ald  = als  + (size_t)GAT_N * 2;           // N*2
  float* mmax = ald  + (size_t)GAT_N * 2;           // N*2
  float* den  = mmax + (size_t)GAT_N * 2;           // N*2
  float* ee   = den  + (size_t)GAT_N * 2;           // Etot*2

  float* out = (float*)d_out;

  // layer 1: 128 -> 2x32 concat, relu
  run_gat_layer(x,   128, W1, as1, ad1, b1, 2, 32, 1, act, ei, E, Etot,
                bufH, agg, als, ald, mmax, den, ee, stream);
  // layer 2: 64 -> 2x32 concat, relu
  run_gat_layer(act,  64, W2, as2, ad2, b2, 2, 32, 1, act, ei, E, Etot,
                bufH, agg, als, ald, mmax, den, ee, stream);
  // layer 3: 64 -> 1x64 (mean over 1 head == identity), no relu -> d_out
  run_gat_layer(act,  64, W3, as3, ad3, b3, 1, 64, 0, out, ei, E, Etot,
                bufH, agg, als, ald, mmax, den, ee, stream);
}